// NeuralKNearestNeighbor_35424890257946
// MI455X (gfx1250) — compile-verified
//
#include <hip/hip_runtime.h>
#include <math.h>

#define B_ 32
#define D_ 256
#define N_ 16384
#define F_ 256
#define K_ 32
#define NSPLIT 8
#define CN 128          // n-chunk staged per iteration in gemm kernel
// TEMP = 0.5 -> 1/t = 2.0

typedef __attribute__((ext_vector_type(2))) float v2f;
typedef __attribute__((ext_vector_type(8))) float v8f;

// ---------------------------------------------------------------------------
// Kernel 1: sims[b,n] = -(query[b,:] . keys[b,:,n]) / sqrt(D)
// Coalesced: 256 threads read 256 consecutive n per d-step. Reads keys 512MB.
// ---------------------------------------------------------------------------
__global__ __launch_bounds__(256)
void nknn_sims_kernel(const float* __restrict__ query,
                      const float* __restrict__ keys,
                      float* __restrict__ sims) {
    __shared__ float qs[D_];
    const int b = blockIdx.y;
    const int n = blockIdx.x * 256 + threadIdx.x;
    qs[threadIdx.x] = query[b * D_ + threadIdx.x];   // D_ == blockDim.x == 256
    __syncthreads();
    const float* kb = keys + (size_t)b * D_ * N_ + n;
    float acc = 0.f;
#pragma unroll 8
    for (int d = 0; d < D_; ++d) {
        acc = fmaf(qs[d], kb[(size_t)d * N_], acc);
    }
    sims[b * N_ + n] = -acc * 0.0625f;   // 1/sqrt(256)
}

// ---------------------------------------------------------------------------
// Block-wide reduction (256 threads = 8 wave32). Deterministic order.
// ---------------------------------------------------------------------------
__device__ __forceinline__ float block_reduce(float v, bool is_max, float* red) {
#pragma unroll
    for (int off = 16; off > 0; off >>= 1) {
        float o = __shfl_down(v, off, 32);
        v = is_max ? fmaxf(v, o) : (v + o);
    }
    __syncthreads();                       // protect red[] from previous use
    if ((threadIdx.x & 31) == 0) red[threadIdx.x >> 5] = v;
    __syncthreads();
    float r = red[0];
#pragma unroll
    for (int w = 1; w < 8; ++w) r = is_max ? fmaxf(r, red[w]) : (r + red[w]);
    return r;
}

// ---------------------------------------------------------------------------
// Kernel 2: per batch, run the full iterative-softmax recurrence with alpha
// held in registers (64 floats per lane x 256 lanes = 16384). Emit the global
// scalars: scal[b][0] = lse of sims softmax; scal[b][1+k] = lse_k of step k.
// ---------------------------------------------------------------------------
__global__ __launch_bounds__(256)
void nknn_lse_kernel(const float* __restrict__ sims,
                     float* __restrict__ scal) {
    __shared__ float red[8];
    const int tid = threadIdx.x;
    const int b = blockIdx.x;
    const float* s = sims + (size_t)b * N_;

    float a[64];
#pragma unroll
    for (int i = 0; i < 64; ++i) a[i] = s[tid + i * 256];

    // alpha0 = softmax(sims)
    float m = -3.4e38f;
#pragma unroll
    for (int i = 0; i < 64; ++i) m = fmaxf(m, a[i]);
    m = block_reduce(m, true, red);
    float ss = 0.f;
#pragma unroll
    for (int i = 0; i < 64; ++i) ss += expf(a[i] - m);
    ss = block_reduce(ss, false, red);
    const float lse_s = m + logf(ss);
    if (tid == 0) scal[b * (K_ + 1)] = lse_s;
#pragma unroll
    for (int i = 0; i < 64; ++i) a[i] = expf(a[i] - lse_s);

    for (int k = 0; k < K_; ++k) {
        float mk = -3.4e38f;
#pragma unroll
        for (int i = 0; i < 64; ++i) mk = fmaxf(mk, a[i]);
        mk = block_reduce(mk, true, red);
        float sk = 0.f;
#pragma unroll
        for (int i = 0; i < 64; ++i) sk += expf(2.f * (a[i] - mk));
        sk = block_reduce(sk, false, red);
        const float lk = 2.f * mk + logf(sk);   // logsumexp of alpha/t
        if (tid == 0) scal[b * (K_ + 1) + 1 + k] = lk;
#pragma unroll
        for (int i = 0; i < 64; ++i) {
            float om = expf(2.f * a[i] - lk);
            a[i] += log1pf(-om);
        }
    }
}

// ---------------------------------------------------------------------------
// Kernel 3: partial[c][b][k][f] = sum over n-chunk of omega[n,k]*values[f,n]
// using V_WMMA_F32_16X16X4_F32. omega recomputed on the fly from sims + lse
// scalars into LDS (padded stride 33 -> conflict-free A-tile reads).
// Wave w: k-tiles {0,1} x f-tiles {2w, 2w+1}; B loaded once per (f-tile,step).
// ---------------------------------------------------------------------------
__global__ __launch_bounds__(256)
void nknn_gemm_kernel(const float* __restrict__ values,
                      const float* __restrict__ sims,
                      const float* __restrict__ scal,
                      float* __restrict__ partial) {
    __shared__ float om[CN][K_ + 1];   // [n][k], padded
    __shared__ float lse[K_ + 1];

    const int b = blockIdx.y;
    const int c = blockIdx.x;
    const int tid = threadIdx.x;
    const int wave = tid >> 5;
    const int lane = tid & 31;
    const int m16 = lane & 15;
    const int half = lane >> 4;
    const int ft0 = wave * 2;                    // two 16-wide f tiles per wave
    const int nBase = c * (N_ / NSPLIT);         // 2048 per split

    if (tid < K_ + 1) lse[tid] = scal[b * (K_ + 1) + tid];

    const float* vb = values + (size_t)b * F_ * N_;
    v8f acc[2][2] = {};                          // [ktile][f subtile]
    __syncthreads();

    for (int ch = 0; ch < (N_ / NSPLIT) / CN; ++ch) {
        const int n0 = nBase + ch * CN;
        __syncthreads();                          // WAR: prior A-reads done
        if (tid < CN) {
            float aa = expf(sims[(size_t)b * N_ + n0 + tid] - lse[0]);
#pragma unroll
            for (int k = 0; k < K_; ++k) {
                float o = expf(2.f * aa - lse[1 + k]);
                om[tid][k] = o;
                aa += log1pf(-o);
            }
        }
        __syncthreads();

        for (int nn = 0; nn < CN; nn += 4) {
            const int na = nn + half * 2;
            // A (16x4 f32): lane m16, K = half*2 + {0,1}; col = kt*16 + m16
            v2f a0, a1;
            a0.x = om[na][m16];       a0.y = om[na + 1][m16];
            a1.x = om[na][16 + m16];  a1.y = om[na + 1][16 + m16];
#pragma unroll
            for (int ftl = 0; ftl < 2; ++ftl) {
                const int f = (ft0 + ftl) * 16 + m16;
                const float* vp = vb + (size_t)f * N_ + n0 + na;
                v2f bb;
                bb.x = vp[0];
                bb.y = vp[1];
                acc[0][ftl] = __builtin_amdgcn_wmma_f32_16x16x4_f32(
                    false, a0, false, bb, (short)0, acc[0][ftl], false, false);
                acc[1][ftl] = __builtin_amdgcn_wmma_f32_16x16x4_f32(
                    false, a1, false, bb, (short)0, acc[1][ftl], false, false);
            }
        }
    }

    // write partial tile: C layout -> VGPR v holds rows v (lanes 0-15) / v+8
    float* pb = partial + ((size_t)c * B_ + b) * (K_ * F_);
#pragma unroll
    for (int kt = 0; kt < 2; ++kt) {
#pragma unroll
        for (int ftl = 0; ftl < 2; ++ftl) {
            const int fcol = (ft0 + ftl) * 16 + m16;
#pragma unroll
            for (int v = 0; v < 8; ++v) {
                const int krow = kt * 16 + v + half * 8;
                pb[krow * F_ + fcol] = acc[kt][ftl][v];
            }
        }
    }
}

// ---------------------------------------------------------------------------
// Kernel 4: deterministic reduction of the NSPLIT partials -> d_out
// ---------------------------------------------------------------------------
__global__ __launch_bounds__(256)
void nknn_reduce_kernel(const float* __restrict__ partial,
                        float* __restrict__ out) {
    const int i = blockIdx.x * 256 + threadIdx.x;   // i = (b*K + k)*F + f
    float s = 0.f;
#pragma unroll
    for (int c = 0; c < NSPLIT; ++c) {
        s += partial[(size_t)c * (B_ * K_ * F_) + i];
    }
    out[i] = s;
}

// ---------------------------------------------------------------------------
extern "C" void kernel_launch(void* const* d_in, const int* in_sizes, int n_in,
                              void* d_out, int out_size, void* d_ws, size_t ws_size,
                              hipStream_t stream) {
    const float* query  = (const float*)d_in[0];   // [B, D]
    const float* keys   = (const float*)d_in[1];   // [B, D, N]
    const float* values = (const float*)d_in[2];   // [B, F, N]
    float* out = (float*)d_out;                    // [B, K, F]

    float* ws      = (float*)d_ws;
    float* sims    = ws;                             // B*N     = 524288 floats
    float* scal    = ws + (size_t)B_ * N_;           // B*(K+1) = 1056 floats
    float* partial = scal + 4096;                    // NSPLIT*B*K*F = 2097152 floats

    nknn_sims_kernel<<<dim3(N_ / 256, B_), 256, 0, stream>>>(query, keys, sims);
    nknn_lse_kernel<<<dim3(B_), 256, 0, stream>>>(sims, scal);
    nknn_gemm_kernel<<<dim3(NSPLIT, B_), 256, 0, stream>>>(values, sims, scal, partial);
    nknn_reduce_kernel<<<dim3((B_ * K_ * F_) / 256), 256, 0, stream>>>(partial, out);
}